// FlowProjectionModule_15393162789085
// MI455X (gfx1250) — compile-verified
//
#include <hip/hip_runtime.h>
#include <math.h>

// Problem shape from the reference: flow (B=8, C=2, H=1080, W=1920) f32.
#define B_  8
#define H_  1080
#define W_  1920
#define HW_ (H_ * W_)          // 2,073,600  (divisible by 4)
#define N_  (B_ * HW_)         // 16,588,800 pixels

// ---------------------------------------------------------------------------
// Pass 1: zero the output planes (acc_x, acc_y live directly in d_out) and the
// hit-count workspace. 128-bit stores; one float4 per thread.
// ---------------------------------------------------------------------------
__global__ void fp_zero_kernel(float4* __restrict__ out4,
                               float4* __restrict__ cnt4,
                               int nOut4, int nCnt4) {
    int q = blockIdx.x * blockDim.x + threadIdx.x;
    const float4 z = make_float4(0.f, 0.f, 0.f, 0.f);
    if (q < nOut4) {
        out4[q] = z;
    } else {
        int r = q - nOut4;
        if (r < nCnt4) cnt4[r] = z;
    }
}

// ---------------------------------------------------------------------------
// Pass 2: scatter. One thread per source pixel. Reads (fx,fy) coalesced,
// computes the 4 clamped corner indices of the landing point, and issues 12
// native f32 atomic adds (unsafeAtomicAdd -> global_atomic_add_f32, which
// RMWs at the L2 atomic units; per-batch accumulator working set ~25MB sits
// inside the 192MB L2). Invalid pixels contribute exact zeros in the
// reference, so they are skipped entirely. global_prefetch_b8 warms the
// accumulator lines near this pixel (atomic targets cluster around the source
// location since flow ~ N(0,5)).
// ---------------------------------------------------------------------------
__global__ void fp_scatter_kernel(const float* __restrict__ flow,
                                  float* __restrict__ out,
                                  float* __restrict__ cnt) {
    int p = blockIdx.x * blockDim.x + threadIdx.x;
    if (p >= N_) return;

    int b   = p / HW_;
    int pix = p - b * HW_;
    int y   = pix / W_;
    int x   = pix - y * W_;

    const float* f = flow + (size_t)b * (2 * HW_);
    float fx = f[pix];
    float fy = f[HW_ + pix];

    float* ox = out + (size_t)b * (2 * HW_);   // acc_x plane of batch b
    float* oy = ox + HW_;                      // acc_y plane of batch b
    float* c  = cnt + (size_t)b * HW_;         // hit-count plane of batch b

    // Warm the L2 lines that the upcoming atomics will hit (targets are
    // within a few rows/cols of the source pixel for this flow magnitude).
    __builtin_prefetch(ox + pix, 1, 1);        // -> global_prefetch_b8
    __builtin_prefetch(oy + pix, 1, 1);
    __builtin_prefetch(c  + pix, 1, 1);

    float x2 = (float)x + fx;
    float y2 = (float)y + fy;

    bool valid = (x2 >= 0.f) && (y2 >= 0.f) &&
                 (x2 <= (float)(W_ - 1)) && (y2 <= (float)(H_ - 1));
    if (!valid) return;   // reference scatters exact zeros here: a no-op

    int xL = (int)floorf(x2);
    int yT = (int)floorf(y2);
    xL = max(0, min(xL, W_ - 1));
    yT = max(0, min(yT, H_ - 1));
    int xR = min(xL + 1, W_ - 1);
    int yB = min(yT + 1, H_ - 1);

    int i00 = yT * W_ + xL;
    int i01 = yT * W_ + xR;
    int i10 = yB * W_ + xL;
    int i11 = yB * W_ + xR;

    float vx = -fx;
    float vy = -fy;

    unsafeAtomicAdd(&ox[i00], vx);
    unsafeAtomicAdd(&ox[i01], vx);
    unsafeAtomicAdd(&ox[i10], vx);
    unsafeAtomicAdd(&ox[i11], vx);

    unsafeAtomicAdd(&oy[i00], vy);
    unsafeAtomicAdd(&oy[i01], vy);
    unsafeAtomicAdd(&oy[i10], vy);
    unsafeAtomicAdd(&oy[i11], vy);

    unsafeAtomicAdd(&c[i00], 1.0f);
    unsafeAtomicAdd(&c[i01], 1.0f);
    unsafeAtomicAdd(&c[i10], 1.0f);
    unsafeAtomicAdd(&c[i11], 1.0f);
}

// ---------------------------------------------------------------------------
// Pass 3: normalize. out = (cnt > 0) ? acc / cnt : 0, done in place on d_out.
// Vectorized: each thread handles 4 pixels (b128 loads/stores).
// ---------------------------------------------------------------------------
__global__ void fp_finalize_kernel(float* __restrict__ out,
                                   const float* __restrict__ cnt) {
    int q = blockIdx.x * blockDim.x + threadIdx.x;
    const int HW4 = HW_ / 4;
    if (q >= N_ / 4) return;

    int b = q / HW4;
    int r = q - b * HW4;

    const float4* cnt4 = (const float4*)cnt;
    float4*       out4 = (float4*)out;

    float4 cv = cnt4[q];                              // cnt is b-major: index q
    size_t ix = (size_t)b * (2 * HW4) + r;            // acc_x tile
    size_t iy = ix + HW4;                             // acc_y tile

    float4 xv = out4[ix];
    float4 yv = out4[iy];

    if (cv.x > 0.f) { xv.x /= cv.x; yv.x /= cv.x; } else { xv.x = 0.f; yv.x = 0.f; }
    if (cv.y > 0.f) { xv.y /= cv.y; yv.y /= cv.y; } else { xv.y = 0.f; yv.y = 0.f; }
    if (cv.z > 0.f) { xv.z /= cv.z; yv.z /= cv.z; } else { xv.z = 0.f; yv.z = 0.f; }
    if (cv.w > 0.f) { xv.w /= cv.w; yv.w /= cv.w; } else { xv.w = 0.f; yv.w = 0.f; }

    out4[ix] = xv;
    out4[iy] = yv;
}

// ---------------------------------------------------------------------------
extern "C" void kernel_launch(void* const* d_in, const int* in_sizes, int n_in,
                              void* d_out, int out_size, void* d_ws, size_t ws_size,
                              hipStream_t stream) {
    const float* flow = (const float*)d_in[0];
    float*       out  = (float*)d_out;          // 2*N_ floats: per-batch [x plane][y plane]
    float*       cnt  = (float*)d_ws;           // N_ floats of scratch (66 MB)

    const int nOut4 = (2 * N_) / 4;
    const int nCnt4 = N_ / 4;
    const int totZ  = nOut4 + nCnt4;

    const int TB = 256;  // 8 wave32s per block

    fp_zero_kernel<<<(totZ + TB - 1) / TB, TB, 0, stream>>>(
        (float4*)out, (float4*)cnt, nOut4, nCnt4);

    fp_scatter_kernel<<<(N_ + TB - 1) / TB, TB, 0, stream>>>(flow, out, cnt);

    fp_finalize_kernel<<<(N_ / 4 + TB - 1) / TB, TB, 0, stream>>>(out, cnt);
}